// Qwen2_5_VLVisionWindowAttention_68135361183882
// MI455X (gfx1250) — compile-verified
//
#include <hip/hip_runtime.h>

typedef float v2f __attribute__((ext_vector_type(2)));
typedef float v4f __attribute__((ext_vector_type(4)));
typedef float v8f __attribute__((ext_vector_type(8)));

#define SEQ      16384
#define DIM      1280
#define NHEADS   16
#define HD       80
#define WIN      64
#define NWIN     (SEQ / WIN)      // 256
#define QKVLD    (3 * DIM)        // 3840
#define QSTR     84               // padded LDS stride for q/k/v tiles (4-float aligned)
#define SSTR     66               // padded (even) LDS stride for score tile

#define WMMA_F32(a, b, c) \
    __builtin_amdgcn_wmma_f32_16x16x4_f32(false, (a), false, (b), (short)0, (c), false, false)

// ---------------------------------------------------------------------------
// C[M,N] = A[M,K] @ W[N,K]^T + bias[N]   (fp32 WMMA 16x16x4)
// One wave computes a 32(M) x 64(N) tile: 8 independent WMMA chains fed by
// 6 b64 loads per K-step, software-pipelined (register double buffering) so
// loads for K+4 are in flight while the WMMAs for K issue.
// Block = 8 waves stacked along M -> block tile 256(M) x 64(N).
// Both weight matrices (19.7MB / 6.5MB) and the activation panel are
// L2-resident (192MB L2), so no LDS staging is needed.
// ---------------------------------------------------------------------------
__global__ __launch_bounds__(256) void gemm_bias_wmma(
    const float* __restrict__ A, const float* __restrict__ W,
    const float* __restrict__ bias, float* __restrict__ C,
    int K, int lda, int ldc)
{
    const int wave = threadIdx.x >> 5;
    const int lane = threadIdx.x & 31;
    const int hl   = lane >> 4;       // half of wave (0/1)
    const int ln   = lane & 15;

    const int m0 = blockIdx.y * 256 + wave * 32;
    const int n0 = blockIdx.x * 64;

    v8f acc[8] = {};   // [2 m-tiles][4 n-tiles]

    const float* pa0 = A + (long)(m0 + ln) * lda + 2 * hl;
    const float* pa1 = pa0 + (long)16 * lda;
    const float* pw0 = W + (long)(n0 + ln) * K + 2 * hl;
    const float* pw1 = pw0 + (long)16 * K;
    const float* pw2 = pw0 + (long)32 * K;
    const float* pw3 = pw0 + (long)48 * K;

    v2f a0 = *(const v2f*)(pa0);
    v2f a1 = *(const v2f*)(pa1);
    v2f b0 = *(const v2f*)(pw0);
    v2f b1 = *(const v2f*)(pw1);
    v2f b2 = *(const v2f*)(pw2);
    v2f b3 = *(const v2f*)(pw3);

    for (int k = 4; k < K; k += 4) {
        // prefetch next K-step while current WMMAs issue
        v2f na0 = *(const v2f*)(pa0 + k);
        v2f na1 = *(const v2f*)(pa1 + k);
        v2f nb0 = *(const v2f*)(pw0 + k);
        v2f nb1 = *(const v2f*)(pw1 + k);
        v2f nb2 = *(const v2f*)(pw2 + k);
        v2f nb3 = *(const v2f*)(pw3 + k);
        acc[0] = WMMA_F32(a0, b0, acc[0]);
        acc[1] = WMMA_F32(a0, b1, acc[1]);
        acc[2] = WMMA_F32(a0, b2, acc[2]);
        acc[3] = WMMA_F32(a0, b3, acc[3]);
        acc[4] = WMMA_F32(a1, b0, acc[4]);
        acc[5] = WMMA_F32(a1, b1, acc[5]);
        acc[6] = WMMA_F32(a1, b2, acc[6]);
        acc[7] = WMMA_F32(a1, b3, acc[7]);
        a0 = na0; a1 = na1;
        b0 = nb0; b1 = nb1; b2 = nb2; b3 = nb3;
    }
    acc[0] = WMMA_F32(a0, b0, acc[0]);
    acc[1] = WMMA_F32(a0, b1, acc[1]);
    acc[2] = WMMA_F32(a0, b2, acc[2]);
    acc[3] = WMMA_F32(a0, b3, acc[3]);
    acc[4] = WMMA_F32(a1, b0, acc[4]);
    acc[5] = WMMA_F32(a1, b1, acc[5]);
    acc[6] = WMMA_F32(a1, b2, acc[6]);
    acc[7] = WMMA_F32(a1, b3, acc[7]);

#pragma unroll
    for (int mi = 0; mi < 2; ++mi) {
#pragma unroll
        for (int t = 0; t < 4; ++t) {
            const int n = n0 + t * 16 + ln;
            const float bv = bias[n];
#pragma unroll
            for (int r = 0; r < 8; ++r) {
                const int m = m0 + mi * 16 + r + hl * 8;
                C[(long)m * ldc + n] = acc[mi * 4 + t][r] + bv;
            }
        }
    }
}

// ---------------------------------------------------------------------------
// Fused windowed attention, one block per (window b, head h):
//   stage q,k,v [64,80] -> LDS (b128 loads), RoPE q/k in LDS (1/sqrt(80)
//   folded into q), scores = q@k^T via WMMA + mask, wave32 shuffle softmax,
//   out = probs@v via WMMA, written into the Q slot of the qkv buffer in
//   merged-head [S, 1280] layout (each block only touches its own region).
// ---------------------------------------------------------------------------
__global__ __launch_bounds__(256) void win_attn_wmma(
    float* __restrict__ qkv,                 // [SEQ, 3840]; q-slot overwritten
    const float* __restrict__ cosb,          // [SEQ, 80]
    const float* __restrict__ sinb,          // [SEQ, 80]
    const float* __restrict__ mask)          // [NWIN, 1, 64, 64]
{
    __shared__ float qs[WIN * QSTR];
    __shared__ float ks[WIN * QSTR];
    __shared__ float vs[WIN * QSTR];
    __shared__ float ss[WIN * SSTR];

    const int b    = blockIdx.x;
    const int h    = blockIdx.y;
    const int tid  = threadIdx.x;
    const int wave = tid >> 5;
    const int lane = tid & 31;
    const int hl   = lane >> 4;
    const int ln   = lane & 15;
    const float scale = 0.1118033988749895f;   // 1/sqrt(80)

    const long base = (long)b * WIN * QKVLD + (long)h * HD;

    // ---- stage q,k,v into LDS: 1280 float4 transfers, fully coalesced ----
    for (int i = tid; i < WIN * (HD / 4); i += 256) {
        const int r = i / (HD / 4), d4 = (i % (HD / 4)) * 4;
        const long g = base + (long)r * QKVLD + d4;
        *(v4f*)&qs[r * QSTR + d4] = *(const v4f*)&qkv[g];
        *(v4f*)&ks[r * QSTR + d4] = *(const v4f*)&qkv[g + DIM];
        *(v4f*)&vs[r * QSTR + d4] = *(const v4f*)&qkv[g + 2 * DIM];
    }
    __syncthreads();

    // ---- RoPE on q (score scale folded) and k ----
    for (int i = tid; i < WIN * (HD / 2); i += 256) {
        const int r = i / (HD / 2), d = i % (HD / 2);       // d in [0,40)
        const int s = b * WIN + r;
        const float c0 = cosb[s * HD + d],      s0 = sinb[s * HD + d];
        const float c1 = cosb[s * HD + d + 40], s1 = sinb[s * HD + d + 40];
        const float q0 = qs[r * QSTR + d], q1 = qs[r * QSTR + d + 40];
        qs[r * QSTR + d]      = (q0 * c0 - q1 * s0) * scale;
        qs[r * QSTR + d + 40] = (q1 * c1 + q0 * s1) * scale;
        const float k0 = ks[r * QSTR + d], k1 = ks[r * QSTR + d + 40];
        ks[r * QSTR + d]      = k0 * c0 - k1 * s0;
        ks[r * QSTR + d + 40] = k1 * c1 + k0 * s1;
    }
    __syncthreads();

    // ---- scores = q @ k^T : each wave owns one A (m) tile and two B (n)
    //      tiles -> shared A fragment, 2 independent WMMA chains ----
    {
        const int mt  = (wave & 3) * 16;
        const int nt0 = (wave >> 2) * 32;
        const int nt1 = nt0 + 16;
        v8f acc0 = {}, acc1 = {};
        for (int kk = 0; kk < HD; kk += 4) {
            v2f a  = *(const v2f*)&qs[(mt  + ln) * QSTR + kk + 2 * hl];
            v2f bA = *(const v2f*)&ks[(nt0 + ln) * QSTR + kk + 2 * hl];
            v2f bB = *(const v2f*)&ks[(nt1 + ln) * QSTR + kk + 2 * hl];
            acc0 = WMMA_F32(a, bA, acc0);
            acc1 = WMMA_F32(a, bB, acc1);
        }
        const float* mwin = mask + (long)b * WIN * WIN;
#pragma unroll
        for (int r = 0; r < 8; ++r) {
            const int row = mt + r + hl * 8;
            ss[row * SSTR + nt0 + ln] = acc0[r] + mwin[row * WIN + nt0 + ln];
            ss[row * SSTR + nt1 + ln] = acc1[r] + mwin[row * WIN + nt1 + ln];
        }
    }
    __syncthreads();

    // ---- row softmax: wave w -> rows 8w..8w+7, lane covers cols {lane, lane+32} ----
    for (int r = 0; r < 8; ++r) {
        const int row = wave * 8 + r;
        float s0 = ss[row * SSTR + lane];
        float s1 = ss[row * SSTR + lane + 32];
        float m = fmaxf(s0, s1);
        for (int off = 16; off > 0; off >>= 1) m = fmaxf(m, __shfl_xor(m, off, 32));
        const float e0 = __expf(s0 - m);
        const float e1 = __expf(s1 - m);
        float sum = e0 + e1;
        for (int off = 16; off > 0; off >>= 1) sum += __shfl_xor(sum, off, 32);
        const float inv = 1.0f / sum;
        ss[row * SSTR + lane]      = e0 * inv;
        ss[row * SSTR + lane + 32] = e1 * inv;
    }
    __syncthreads();

    // ---- out = probs @ v : 4(M) x 5(N) tiles, K = 64; write into Q slot ----
    for (int t = wave; t < 20; t += 8) {
        const int mt = (t / 5) * 16, nt = (t % 5) * 16;
        v8f acc = {};
        for (int kk = 0; kk < WIN; kk += 4) {
            v2f a = *(const v2f*)&ss[(mt + ln) * SSTR + kk + 2 * hl];
            v2f bb;
            bb.x = vs[(kk + 2 * hl)     * QSTR + nt + ln];
            bb.y = vs[(kk + 2 * hl + 1) * QSTR + nt + ln];
            acc = WMMA_F32(a, bb, acc);
        }
#pragma unroll
        for (int r = 0; r < 8; ++r) {
            const int row = mt + r + hl * 8, d = nt + ln;
            qkv[base + (long)row * QKVLD + d] = acc[r];   // merged-head output
        }
    }
}

extern "C" void kernel_launch(void* const* d_in, const int* in_sizes, int n_in,
                              void* d_out, int out_size, void* d_ws, size_t ws_size,
                              hipStream_t stream) {
    const float* hidden = (const float*)d_in[0];
    const float* mask   = (const float*)d_in[1];
    const float* cosb   = (const float*)d_in[2];
    const float* sinb   = (const float*)d_in[3];
    const float* qkv_w  = (const float*)d_in[4];
    const float* qkv_b  = (const float*)d_in[5];
    const float* proj_w = (const float*)d_in[6];
    const float* proj_b = (const float*)d_in[7];
    float* out = (float*)d_out;
    float* qkv = (float*)d_ws;           // [SEQ, 3840] fp32 intermediate

    // 1) fused QKV projection: [16384,1280] @ [3840,1280]^T + b -> ws
    gemm_bias_wmma<<<dim3(QKVLD / 64, SEQ / 256), 256, 0, stream>>>(
        hidden, qkv_w, qkv_b, qkv, DIM, DIM, QKVLD);

    // 2) fused RoPE + windowed attention; output lands in the Q slot of ws
    win_attn_wmma<<<dim3(NWIN, NHEADS), 256, 0, stream>>>(qkv, cosb, sinb, mask);

    // 3) output projection: [16384,1280](lda=3840) @ [1280,1280]^T + b -> out
    gemm_bias_wmma<<<dim3(DIM / 64, SEQ / 256), 256, 0, stream>>>(
        qkv, proj_w, proj_b, out, DIM, QKVLD, DIM);
}